// DiTAC_69647189672226
// MI455X (gfx1250) — compile-verified
//
#include <hip/hip_runtime.h>
#include <math.h>

// ---------------- DiTAC constants ----------------
#define NC      10
#define NTHETA  9        // NC-1
#define TWO_NC  20
#define A_LO    (-0.5f)
#define B_HI    (2.0f)
#define INV_RANGE 0.4f       // 1/(B_HI-A_LO)
#define RANGE     2.5f
#define INV_SQRT2 0.70710678118654752f

struct Bmat { float v[TWO_NC * NTHETA]; };   // row-major [20][9], kernarg by value

typedef __attribute__((ext_vector_type(2))) float v2f;
typedef __attribute__((ext_vector_type(8))) float v8f;

// ---------------------------------------------------------------------------
// Main kernel: per-block WMMA GEMV (A = Bnull @ theta) -> LDS cell table,
// then float4 grid-stride CPAB + erf-gate streaming loop.
// ---------------------------------------------------------------------------
__global__ __launch_bounds__(256) void ditac_kernel(
    const float* __restrict__ xin,
    const float* __restrict__ theta,
    float* __restrict__ out,
    int n, Bmat Bm)
{
    __shared__ __align__(16) float  sA[24];       // raw A cells: a0,b0,a1,b1,...
    __shared__ __align__(16) float4 sC[NC];       // {a, b, b/a_safe, 1/a_safe}

    const int lane = threadIdx.x;

    // ---- wave 0: A = Bnull(20x9) @ theta via V_WMMA_F32_16X16X4_F32 ----
    if (lane < 32) {
        const bool hi   = lane >= 16;
        const int  mloc = lane & 15;
        #pragma unroll
        for (int g = 0; g < 2; ++g) {             // row groups 0..15, 16..19(pad)
            v8f acc = {0.f, 0.f, 0.f, 0.f, 0.f, 0.f, 0.f, 0.f};
            #pragma unroll
            for (int j = 0; j < 3; ++j) {         // K chunks {0..3},{4..7},{8,pad}
                const int k0 = 4 * j + (hi ? 2 : 0);
                const int k1 = k0 + 1;
                const int M  = 16 * g + mloc;
                v2f av, bv;
                // A-frag 16x4 f32: lane = M%16; lo lanes K={4j,4j+1}, hi lanes K={4j+2,4j+3}
                av.x = (M < TWO_NC && k0 < NTHETA) ? Bm.v[M * NTHETA + k0] : 0.f;
                av.y = (M < TWO_NC && k1 < NTHETA) ? Bm.v[M * NTHETA + k1] : 0.f;
                // B-frag 4x16 f32, only column N=0 nonzero (theta): rows K live at lanes 0/16
                float b0 = 0.f, b1 = 0.f;
                if (mloc == 0) {
                    if (k0 < NTHETA) b0 = theta[k0];
                    if (k1 < NTHETA) b1 = theta[k1];
                }
                bv.x = b0; bv.y = b1;
                acc = __builtin_amdgcn_wmma_f32_16x16x4_f32(
                          false, av, false, bv, (short)0, acc, false, false);
            }
            // D column 0: lane 0 holds M=r in acc[r], lane 16 holds M=r+8
            if (lane == 0) {
                #pragma unroll
                for (int r = 0; r < 8; ++r) {
                    const int idx = 16 * g + r;
                    if (idx < TWO_NC) sA[idx] = acc[r];
                }
            }
            if (lane == 16 && g == 0) {
                #pragma unroll
                for (int r = 0; r < 8; ++r) sA[8 + r] = acc[r];
            }
        }
        // per-cell invariants (same wave; LDS ops are in-order within a wave)
        if (lane < NC) {
            const float a   = sA[2 * lane];
            const float b   = sA[2 * lane + 1];
            const bool  big = fabsf(a) > 1e-8f;
            const float as  = big ? a : 1.0f;
            sC[lane] = make_float4(a, b, b / as, 1.0f / as);
        }
    }
    __syncthreads();

    // ---- streaming elementwise loop (float4 vectors) ----
    const int tid    = blockIdx.x * blockDim.x + threadIdx.x;
    const int stride = gridDim.x * blockDim.x;
    const int n4     = n >> 2;
    const float4* __restrict__ in4  = (const float4*)xin;
    float4* __restrict__       out4 = (float4*)out;

    for (int i = tid; i < n4; i += stride) {
        const int inext = i + stride;
        if (inext < n4) __builtin_prefetch((const void*)(in4 + inext), 0, 0);

        const float4 vin = in4[i];
        float f[4] = { vin.x, vin.y, vin.z, vin.w };
        bool  mask[4]; float x[4], t[4]; int c[4]; bool done[4];

        #pragma unroll
        for (int e = 0; e < 4; ++e) {
            mask[e] = (f[e] >= A_LO) && (f[e] <= B_HI);
            const float m  = mask[e] ? f[e] : 0.f;
            const float xn = (m - A_LO) * INV_RANGE;
            c[e]    = min(max((int)floorf(xn * (float)NC), 0), NC - 1);
            x[e]    = xn;
            t[e]    = 1.f;
            done[e] = false;
        }

        // CPAB: fixed-point iteration; 'inside' is sticky, so wave-uniform
        // early exit is bit-exact vs. the reference's 11 iterations.
        for (int it = 0; it < NC + 1; ++it) {
            #pragma unroll
            for (int e = 0; e < 4; ++e) {
                const float4 tb = sC[c[e]];
                const float a = tb.x, b = tb.y, r = tb.z, ia = tb.w;
                const bool  big = fabsf(a) > 1e-8f;
                const float v   = fmaf(a, x[e], b);
                const float ea  = __expf(t[e] * a);
                const float psi = big ? fmaf(ea, x[e], r * (ea - 1.f))
                                      : fmaf(t[e], b, x[e]);
                const float left  = (float)c[e] * (1.0f / NC);
                const float right = left + (1.0f / NC);
                const bool  inside = (psi >= left) && (psi <= right);
                const float xb = (v >= 0.f) ? right : left;
                const float vb = fmaf(a, xb, b);
                const float vs = (fabsf(v) > 1e-12f) ? v
                                 : ((v >= 0.f) ? 1e-12f : -1e-12f);
                const float t_aff = __logf(fmaxf(__fdividef(vb, vs), 1e-12f)) * ia;
                const float bs    = (fabsf(b) > 1e-12f) ? b : 1e-12f;
                const float t_hit = big ? t_aff : __fdividef(xb - x[e], bs);
                x[e] = inside ? psi : xb;
                t[e] = inside ? 0.f : fmaxf(t[e] - t_hit, 0.f);
                const int step = (v >= 0.f) ? 1 : -1;
                c[e] = inside ? c[e] : min(max(c[e] + step, 0), NC - 1);
                done[e] = inside;
            }
            if (__all(done[0] && done[1] && done[2] && done[3])) break;
        }

        float4 o;
        float* op = &o.x;
        #pragma unroll
        for (int e = 0; e < 4; ++e) {
            const float data_t = fmaf(x[e], RANGE, A_LO);
            const float cp  = mask[e] ? data_t : f[e];
            const float phi = 0.5f * (1.0f + erff(f[e] * INV_SQRT2));
            op[e] = cp * phi;
        }
        out4[i] = o;
    }

    // ---- scalar tail (n % 4) ----
    for (int i = (n4 << 2) + tid; i < n; i += stride) {
        const float fe   = xin[i];
        const bool  mask = (fe >= A_LO) && (fe <= B_HI);
        const float m    = mask ? fe : 0.f;
        float xe = (m - A_LO) * INV_RANGE;
        int   ce = min(max((int)floorf(xe * (float)NC), 0), NC - 1);
        float te = 1.f;
        for (int it = 0; it < NC + 1; ++it) {
            const float4 tb = sC[ce];
            const float a = tb.x, b = tb.y, r = tb.z, ia = tb.w;
            const bool  big = fabsf(a) > 1e-8f;
            const float v   = fmaf(a, xe, b);
            const float ea  = __expf(te * a);
            const float psi = big ? fmaf(ea, xe, r * (ea - 1.f)) : fmaf(te, b, xe);
            const float left  = (float)ce * (1.0f / NC);
            const float right = left + (1.0f / NC);
            const bool  inside = (psi >= left) && (psi <= right);
            const float xb = (v >= 0.f) ? right : left;
            const float vb = fmaf(a, xb, b);
            const float vs = (fabsf(v) > 1e-12f) ? v : ((v >= 0.f) ? 1e-12f : -1e-12f);
            const float t_aff = __logf(fmaxf(__fdividef(vb, vs), 1e-12f)) * ia;
            const float bs    = (fabsf(b) > 1e-12f) ? b : 1e-12f;
            const float t_hit = big ? t_aff : __fdividef(xb - xe, bs);
            xe = inside ? psi : xb;
            te = inside ? 0.f : fmaxf(te - t_hit, 0.f);
            const int step = (v >= 0.f) ? 1 : -1;
            ce = inside ? ce : min(max(ce + step, 0), NC - 1);
        }
        const float data_t = fmaf(xe, RANGE, A_LO);
        const float cp  = mask ? data_t : fe;
        const float phi = 0.5f * (1.0f + erff(fe * INV_SQRT2));
        out[i] = cp * phi;
    }
}

// ---------------------------------------------------------------------------
// Host: reproduce numpy.linalg.qr(L.T, mode='complete')[:, nc+1:]
// with LAPACK-convention Householder reflectors (dlarfg sign rule).
// Deterministic, recomputed every call (no static state).
// ---------------------------------------------------------------------------
static void compute_basis(Bmat* out)
{
    double L[NC + 1][TWO_NC];
    for (int r = 0; r < NC + 1; ++r)
        for (int cc = 0; cc < TWO_NC; ++cc) L[r][cc] = 0.0;
    for (int i = 1; i < NC; ++i) {
        const double xi = (double)i / NC;
        L[i - 1][2 * (i - 1)]     =  xi;
        L[i - 1][2 * (i - 1) + 1] =  1.0;
        L[i - 1][2 * i]           = -xi;
        L[i - 1][2 * i + 1]       = -1.0;
    }
    L[NC - 1][1]                 = 1.0;
    L[NC][2 * (NC - 1)]          = 1.0;
    L[NC][2 * (NC - 1) + 1]      = 1.0;

    // A = L^T : 20 x 11
    const int M = TWO_NC, N = NC + 1;
    double A[TWO_NC][NC + 1];
    for (int r = 0; r < M; ++r)
        for (int cc = 0; cc < N; ++cc) A[r][cc] = L[cc][r];

    double V[NC + 1][TWO_NC];
    double tau[NC + 1];
    for (int k = 0; k < N; ++k) {
        tau[k] = 0.0;
        for (int i = 0; i < M; ++i) V[k][i] = 0.0;
    }
    for (int k = 0; k < N; ++k) {
        double nrm2 = 0.0;
        for (int i = k; i < M; ++i) nrm2 += A[i][k] * A[i][k];
        const double nrm = sqrt(nrm2);
        if (nrm < 1e-300) continue;
        const double alpha = A[k][k];
        const double beta  = (alpha >= 0.0) ? -nrm : nrm;  // LAPACK dlarfg sign
        tau[k] = (beta - alpha) / beta;
        const double v0 = alpha - beta;
        V[k][k] = 1.0;
        for (int i = k + 1; i < M; ++i) V[k][i] = A[i][k] / v0;
        for (int j = k + 1; j < N; ++j) {
            double s = A[k][j];
            for (int i = k + 1; i < M; ++i) s += V[k][i] * A[i][j];
            s *= tau[k];
            A[k][j] -= s;
            for (int i = k + 1; i < M; ++i) A[i][j] -= V[k][i] * s;
        }
        A[k][k] = beta;
    }
    // Q = H0 H1 ... H10 applied to identity (dorgqr equivalent)
    double Q[TWO_NC][TWO_NC];
    for (int r = 0; r < M; ++r)
        for (int cc = 0; cc < M; ++cc) Q[r][cc] = (r == cc) ? 1.0 : 0.0;
    for (int k = N - 1; k >= 0; --k) {
        if (tau[k] == 0.0) continue;
        for (int j = 0; j < M; ++j) {
            double s = Q[k][j];
            for (int i = k + 1; i < M; ++i) s += V[k][i] * Q[i][j];
            s *= tau[k];
            Q[k][j] -= s;
            for (int i = k + 1; i < M; ++i) Q[i][j] -= V[k][i] * s;
        }
    }
    for (int r = 0; r < TWO_NC; ++r)
        for (int kk = 0; kk < NTHETA; ++kk)
            out->v[r * NTHETA + kk] = (float)Q[r][N + kk];
}

extern "C" void kernel_launch(void* const* d_in, const int* in_sizes, int n_in,
                              void* d_out, int out_size, void* d_ws, size_t ws_size,
                              hipStream_t stream)
{
    const float* x     = (const float*)d_in[0];
    const float* theta = (const float*)d_in[1];
    float*       out   = (float*)d_out;
    const int n = in_sizes[0];

    Bmat Bm;
    compute_basis(&Bm);

    const int threads = 256;
    const int n4 = n >> 2;
    int blocks = (n4 + threads - 1) / threads;
    if (blocks > 4096) blocks = 4096;
    if (blocks < 1)    blocks = 1;

    ditac_kernel<<<blocks, threads, 0, stream>>>(x, theta, out, n, Bm);
}